// GCNConv_58901181497715
// MI455X (gfx1250) — compile-verified
//
#include <hip/hip_runtime.h>
#include <math.h>
#include <stdint.h>

// Problem constants (match reference).
#define NROW 8192
#define DIN  128
#define DOUT 128

typedef __attribute__((ext_vector_type(2))) float v2f;
typedef __attribute__((ext_vector_type(4))) float v4f;
typedef __attribute__((ext_vector_type(8))) float v8f;

// Generic LDS pointer -> 32-bit LDS offset (ISA 10.2: LDS addr = flat[31:0]).
__device__ __forceinline__ unsigned lds_off(const void* p) {
    return (unsigned)(uintptr_t)p;
}

// ---------------------------------------------------------------------------
// Kernel 1: deterministic partial column sums of A (fixed order per thread).
// grid = (NROW/1024, 64), block = 256; float4 coalesced, non-temporal (A is
// a single-pass 256 MB stream; keep it out of L2).
// ---------------------------------------------------------------------------
__global__ __launch_bounds__(256)
void colsum_partial(const float* __restrict__ A, float* __restrict__ part) {
    const int c4 = (blockIdx.x * 256 + threadIdx.x) * 4;
    const int r0 = blockIdx.y * 128;
    v4f s = {};
    const float* p = A + (size_t)r0 * NROW + c4;
#pragma unroll 8
    for (int i = 0; i < 128; ++i) {
        const v4f v = __builtin_nontemporal_load((const v4f*)(p + (size_t)i * NROW));
        s += v;
    }
    *(v4f*)(part + (size_t)blockIdx.y * NROW + c4) = s;
}

// ---------------------------------------------------------------------------
// Kernel 2: d[j] = (colsum_j + 1)^(-1/2); fixed-order reduction (deterministic).
// ---------------------------------------------------------------------------
__global__ __launch_bounds__(256)
void finalize_d(const float* __restrict__ part, float* __restrict__ d) {
    const int j = blockIdx.x * 256 + threadIdx.x;
    float s = 1.0f;                                // identity diagonal term
    for (int c = 0; c < 64; ++c) s += part[(size_t)c * NROW + j];
    d[j] = 1.0f / sqrtf(s);
}

// ---------------------------------------------------------------------------
// WMMA fp32 GEMM:  out[m,:] = post( scale[m] * ( (A @ B)[m,:] + (ADD_I ? B[m,:] : 0) ) )
//   A: M x K row-major (lda), B: K x 128 row-major. The "+B[m,:]" epilogue term
//   implements (A+I)@B without touching the A stream.
// M-tile 64, N=128, K-tile 32; block = 128 threads = 4 waves; wave w owns rows
// [16w,16w+16) and all 8 N-frags (8 x v8f accumulators).
// A tiles: GLOBAL_LOAD_ASYNC_TO_LDS_B128, double-buffered, non-temporal.
// B tiles: register-prefetched into a fragment-ready interleaved layout
//   Bs[p][c] = float2{ B[2p][c], B[2p+1][c] }, pair-row stride 288 dwords
//   -> each B fragment is one bank-conflict-free ds_load_b64.
// ---------------------------------------------------------------------------
template <bool ADD_I, bool RELU>
__global__ __launch_bounds__(128)
void gemm_rowscale(const float* __restrict__ A, int lda, int K,
                   const float* __restrict__ B,
                   const float* __restrict__ scale,
                   float* __restrict__ out) {
    constexpr int MT = 64, KT = 32;
    constexpr int LDA_S = 36;          // dwords per A-tile row (64-bank friendly pad)
    constexpr int ASZ   = MT * LDA_S;  // 2304 dwords per A buffer
    constexpr int LDB_S = 288;         // dwords per B pair-row (2*128 + 32 pad)
    constexpr int BSZ   = 16 * LDB_S;  // 4608 dwords per B buffer
    __shared__ float As[2 * ASZ];
    __shared__ float Bs[2 * BSZ];

    const int tid  = threadIdx.x;
    const int lane = tid & 31;
    const int wv   = tid >> 5;     // 0..3
    const int l16  = lane & 15;
    const int lh   = lane >> 4;    // half-wave select (ISA 7.12.2 layouts)
    const int m0   = blockIdx.x * MT;

    v8f acc[8];
#pragma unroll
    for (int i = 0; i < 8; ++i) acc[i] = {};

    // ---- A tile: 4 async b128 copies per thread (512 float4 slots / 128 thr)
    auto issueA = [&](int k0, int nb) {
#pragma unroll
        for (int it = 0; it < 4; ++it) {
            const int idx = it * 128 + tid;
            const int r = idx >> 3, c4 = (idx & 7) << 2;
            const float* g = A + (size_t)(m0 + r) * lda + (k0 + c4);
            const unsigned l = lds_off(&As[nb * ASZ + r * LDA_S + c4]);
            asm volatile("global_load_async_to_lds_b128 %0, %1, off th:TH_LOAD_NT"
                         :: "v"(l), "v"(g) : "memory");
        }
    };

    // ---- B tile: register prefetch (issued before compute, stored after)
    float2 rb0[8], rb1[8];
    auto loadB = [&](int k0) {
#pragma unroll
        for (int it = 0; it < 8; ++it) {
            const int idx = it * 128 + tid;
            const int p = idx >> 6, c2 = idx & 63;
            const float* g = B + (size_t)(k0 + 2 * p) * DOUT + 2 * c2;
            rb0[it] = *(const float2*)g;
            rb1[it] = *(const float2*)(g + DOUT);
        }
    };
    auto storeB = [&](int nb) {
#pragma unroll
        for (int it = 0; it < 8; ++it) {
            const int idx = it * 128 + tid;
            const int p = idx >> 6, c2 = idx & 63;
            v4f v; v.x = rb0[it].x; v.y = rb1[it].x; v.z = rb0[it].y; v.w = rb1[it].y;
            *(v4f*)&Bs[nb * BSZ + p * LDB_S + 4 * c2] = v;      // 16B aligned
        }
    };

    // ---- compute one K-tile from LDS buffer nb
    auto compute = [&](int nb) {
        const float* as = &As[nb * ASZ + (wv * 16 + l16) * LDA_S + 2 * lh];
        const float* bs = &Bs[nb * BSZ + lh * LDB_S + 2 * l16];
#pragma unroll
        for (int kk = 0; kk < KT; kk += 4) {
            const v2f a = *(const v2f*)(as + kk);             // A frag {K,K+1}
            const float* bk = bs + (kk >> 1) * LDB_S;
#pragma unroll
            for (int nf = 0; nf < 8; ++nf) {
                const v2f b = *(const v2f*)(bk + nf * 32);    // B frag {K,K+1}
                acc[nf] = __builtin_amdgcn_wmma_f32_16x16x4_f32(
                    false, a, false, b, (short)0, acc[nf], false, false);
            }
        }
    };

    // ---- software pipeline: tile t+1 streams while tile t computes
    const int T = K / KT;
    issueA(0, 0);
    loadB(0);
    storeB(0);
    asm volatile("s_wait_asynccnt 0x0" ::: "memory");
    __syncthreads();

    for (int t = 0; t < T; ++t) {
        const int cur = t & 1, nxt = cur ^ 1;
        const bool hn = (t + 1) < T;
        if (hn) { issueA((t + 1) * KT, nxt); loadB((t + 1) * KT); }
        compute(cur);
        if (hn) storeB(nxt);
        asm volatile("s_wait_asynccnt 0x0" ::: "memory");
        __syncthreads();
    }

    // ---- epilogue: (+ B[row,:] if ADD_I), per-row scale, ReLU, store
#pragma unroll
    for (int g = 0; g < 8; ++g) {
        const int row = m0 + wv * 16 + g + 8 * lh;     // C/D layout (ISA 7.12.2)
        const float s = scale[row];
#pragma unroll
        for (int nf = 0; nf < 8; ++nf) {
            float v = acc[nf][g];
            if (ADD_I) v += B[(size_t)row * DOUT + nf * 16 + l16];
            v *= s;
            if (RELU) v = fmaxf(v, 0.f);
            out[(size_t)row * DOUT + nf * 16 + l16] = v;
        }
    }
}

// ---------------------------------------------------------------------------
// Launch: inputs {X [8192x128], A [8192x8192], W [128x128]}, all fp32.
// ws layout: part (64*8192 f) | d (8192 f) | Y (8192*128 f)  ~= 6.3 MB.
// ---------------------------------------------------------------------------
extern "C" void kernel_launch(void* const* d_in, const int* in_sizes, int n_in,
                              void* d_out, int out_size, void* d_ws, size_t ws_size,
                              hipStream_t stream) {
    const float* X = (const float*)d_in[0];
    const float* A = (const float*)d_in[1];
    const float* W = (const float*)d_in[2];
    float* out  = (float*)d_out;
    float* part = (float*)d_ws;            // 64 * NROW
    float* dvec = part + (size_t)64 * NROW;
    float* Y    = dvec + NROW;             // NROW * DOUT

    // 1) d = colsum(A+I)^-1/2
    colsum_partial<<<dim3(NROW / 1024, 64), 256, 0, stream>>>(A, part);
    finalize_d<<<NROW / 256, 256, 0, stream>>>(part, dvec);

    // 2) Y = diag(d) * (X @ W)          (K = 128)
    gemm_rowscale<false, false><<<NROW / 64, 128, 0, stream>>>(X, DIN, DIN, W, dvec, Y);

    // 3) out = relu(diag(d) * ((A+I) @ Y))   (K = 8192, +I via epilogue)
    gemm_rowscale<true, true><<<NROW / 64, 128, 0, stream>>>(A, NROW, NROW, Y, dvec, out);
}